// PolynomialFeatures_50044958933546
// MI455X (gfx1250) — compile-verified
//
#include <hip/hip_runtime.h>

// ---------------------------------------------------------------------------
// PolynomialFeatures(degree=3, include_bias) on [65536 x 16] fp32.
// Store-bound: ~254 MB out vs ~116 MFLOP -> floor ~11us at 23.3 TB/s.
// CDNA5 path: TDM (tensor_load_to_lds, TENSORcnt) stages the x-tile into LDS;
// per-thread loop-invariant term descriptors (one b128 constant load);
// LDS broadcast gathers; fully coalesced non-temporal b32 stores.
// ---------------------------------------------------------------------------

#define POLY_F        16
#define POLY_TERMS    969          // 1 + 16 + 136 + 816
#define POLY_BLOCK    256
#define POLY_QPT      4            // columns per thread (4*256 = 1024 >= 969)
#define POLY_SPB      64           // samples per block

typedef unsigned int u32;
typedef __attribute__((ext_vector_type(4))) unsigned int u32x4;
typedef __attribute__((ext_vector_type(4))) int          i32x4;
typedef __attribute__((ext_vector_type(8))) int          i32x8;

// ---- compile-time term table ----------------------------------------------
// Each term packs (i, j, k) feature indices, sentinel 16 => multiply by 1.0.
// Column order matches the reference: [bias | deg1 | deg2 cwr | deg3 cwr].
// Layout is interleaved so thread t's 4 columns {t, t+256, t+512, t+768}
// are one 16B-aligned u32x4 -> single global_load_b128 per thread.
struct alignas(16) PackedTable { u32 v[POLY_BLOCK][POLY_QPT]; };

constexpr PackedTable makeTable() {
  u32 raw[POLY_BLOCK * POLY_QPT] = {};
  int n = 0;
  constexpr u32 SENT = 16u | (16u << 8) | (16u << 16);
  raw[n++] = SENT;                                                  // bias
  for (u32 i = 0; i < 16; ++i)                                      // degree 1
    raw[n++] = i | (16u << 8) | (16u << 16);
  for (u32 i = 0; i < 16; ++i)                                      // degree 2
    for (u32 j = i; j < 16; ++j)
      raw[n++] = i | (j << 8) | (16u << 16);
  for (u32 i = 0; i < 16; ++i)                                      // degree 3
    for (u32 j = i; j < 16; ++j)
      for (u32 k = j; k < 16; ++k)
        raw[n++] = i | (j << 8) | (k << 16);
  for (; n < POLY_BLOCK * POLY_QPT; ++n) raw[n] = SENT;             // pad

  PackedTable p{};
  for (int t = 0; t < POLY_BLOCK; ++t)
    for (int q = 0; q < POLY_QPT; ++q)
      p.v[t][q] = raw[q * POLY_BLOCK + t];
  return p;
}

__device__ __constant__ PackedTable g_ptbl = makeTable();

#if __has_builtin(__builtin_amdgcn_tensor_load_to_lds) && \
    __has_builtin(__builtin_amdgcn_s_wait_tensorcnt)
#define POLY_USE_TDM 1
#endif

__global__ __launch_bounds__(POLY_BLOCK)
void PolynomialFeatures_kernel(const float* __restrict__ x,
                               float* __restrict__ out) {
  __shared__ __align__(16) float xs[POLY_SPB * POLY_F];   // 4 KB sample tile

  const u32 tid = threadIdx.x;
  const u32 blk = blockIdx.x;

  // One b128 load: this thread's 4 loop-invariant term descriptors.
  const u32x4 e4 = *(const u32x4*)&g_ptbl.v[tid][0];

  const float* gx = x + (size_t)blk * (POLY_SPB * POLY_F);

#ifdef POLY_USE_TDM
  // ---- Tensor Data Mover: DMA the 4 KB contiguous tile global -> LDS ------
  if ((tid >> 5) == 0) {                       // wave 0 issues the descriptor
    unsigned long long ga = (unsigned long long)(size_t)gx;   // 57-bit VA
    u32 ldsOff = (u32)(size_t)(&xs[0]);        // low 32 bits = LDS byte offset
    constexpr u32 E8 = (POLY_SPB * POLY_F * 4) / 8;           // 512 x 8B elems

    // D# group 0: count=1 | lds_addr | global_addr[56:0] | type=2
    u32x4 g0;
    g0[0] = 1u;
    g0[1] = ldsOff;
    g0[2] = (u32)(ga & 0xFFFFFFFFull);
    g0[3] = (u32)((ga >> 32) & 0x01FFFFFFull) | (2u << 30);

    // D# group 1: wg_mask=0, data_size=3 (8B), 1-D tile of E8 elements
    i32x8 g1;
    g1[0] = (int)(3u << 16);                            // data_size = 8 bytes
    g1[1] = (int)((E8 & 0xFFFFu) << 16);                // tensor_dim0[15:0]
    g1[2] = (int)(((E8 >> 16) & 0xFFFFu) | (1u << 16)); // dim0 hi | tensor_dim1=1
    g1[3] = (int)((E8 & 0xFFFFu) << 16);                // tile_dim0 = E8
    g1[4] = 1;                                          // tile_dim1=1, tile_dim2=0
    g1[5] = (int)E8;                                    // tensor_dim0_stride lo
    g1[6] = 0;
    g1[7] = 0;

    i32x4 g2 = (i32x4)0;
    i32x4 g3 = (i32x4)0;
#if __clang_major__ >= 23
    i32x8 g4 = (i32x8)0;
    __builtin_amdgcn_tensor_load_to_lds(g0, g1, g2, g3, g4, 0);
#else
    __builtin_amdgcn_tensor_load_to_lds(g0, g1, g2, g3, 0);
#endif
    __builtin_amdgcn_s_wait_tensorcnt(0);
  }
#else
  // ---- fallback: vectorized synchronous staging ---------------------------
  {
    const float4* src = (const float4*)gx;
    float4*       dst = (float4*)xs;
    for (u32 t = tid; t < (POLY_SPB * POLY_F) / 4; t += POLY_BLOCK)
      dst[t] = src[t];
  }
#endif
  __syncthreads();

  // ---- main loop: fixed columns per thread, samples outer ------------------
  float* ob = out + (size_t)blk * (POLY_SPB * POLY_TERMS);

  // q == 3 covers columns 768..1023; only 768..968 are real.
  const bool q3ok = tid < (POLY_TERMS - 3 * POLY_BLOCK);   // tid < 201

#pragma unroll 2
  for (u32 s = 0; s < POLY_SPB; ++s) {
    const float* row  = &xs[s * POLY_F];     // all lanes hit one 16-float row
    float*       orow = ob + s * POLY_TERMS;

#pragma unroll
    for (u32 q = 0; q < POLY_QPT; ++q) {
      const u32 e = e4[q];
      const u32 i =  e        & 0xFFu;       // loop-invariant per thread:
      const u32 j = (e >> 8)  & 0xFFu;       // cmp masks hoisted out of the
      const u32 k = (e >> 16) & 0xFFu;       // sample loop by the compiler

      float a = row[i & 15u];
      float b = row[j & 15u];
      float c = row[k & 15u];
      a = (i < 16u) ? a : 1.0f;              // sentinel -> 1.0
      b = (j < 16u) ? b : 1.0f;
      c = (k < 16u) ? c : 1.0f;

      // lanes 0..255 write consecutive dwords: fully coalesced NT store
      if (q < 3u || q3ok)
        __builtin_nontemporal_store(a * b * c, orow + q * POLY_BLOCK + tid);
    }
  }
}

extern "C" void kernel_launch(void* const* d_in, const int* in_sizes, int n_in,
                              void* d_out, int out_size, void* d_ws, size_t ws_size,
                              hipStream_t stream) {
  (void)n_in; (void)out_size; (void)d_ws; (void)ws_size;
  const float* x   = (const float*)d_in[0];
  float*       out = (float*)d_out;

  const int n    = in_sizes[0] / POLY_F;   // 65536
  const int grid = n / POLY_SPB;           // 1024 blocks

  PolynomialFeatures_kernel<<<grid, POLY_BLOCK, 0, stream>>>(x, out);
}